// GINEBlock_49323404427797
// MI455X (gfx1250) — compile-verified
//
#include <hip/hip_runtime.h>
#include <hip/hip_bf16.h>
#include <float.h>

// ---------------------------------------------------------------------------
// GINE block on gfx1250: fp32 WMMA (v_wmma_f32_16x16x4_f32) GEMMs fused with
// gather / scatter-atomic aggregation. Edge pass uses double-buffered
// GLOBAL_LOAD_ASYNC_TO_LDS_B128 (ASYNCcnt) to overlap the edge_attr stream
// with the WMMA chain. f64 LayerNorm statistics fused into the node pass.
// ---------------------------------------------------------------------------

typedef __attribute__((ext_vector_type(2))) float v2f;
typedef __attribute__((ext_vector_type(4))) float v4f;
typedef __attribute__((ext_vector_type(8))) float v8f;

#define DD   128
#define LDA  132           // 128 + 4 pad -> banks (4*m + c) % 64, conflict-free

// Async global->LDS copy, 16B per lane (gfx1250, tracked by ASYNCcnt).
__device__ __forceinline__ void async_ld_b128(void* lds, const float* g) {
    asm volatile("global_load_async_to_lds_b128 %0, %1, off"
                 :: "v"((unsigned)(size_t)lds),
                    "v"((unsigned long long)(size_t)g)
                 : "memory");
}
__device__ __forceinline__ void wait_async0() {
    asm volatile("s_wait_asynccnt 0x0" ::: "memory");
}

// -------------------------------- init ------------------------------------
__global__ void k_zero(float* __restrict__ agg, double* __restrict__ sums, int total4) {
    int i = blockIdx.x * blockDim.x + threadIdx.x;
    if (i == 0) { sums[0] = 0.0; sums[1] = 0.0; }
    v4f z = {0.f, 0.f, 0.f, 0.f};
    for (; i < total4; i += gridDim.x * blockDim.x)
        ((v4f*)agg)[i] = z;
}

// ------------------------- edge pass (GEMM + scatter) ----------------------
// tile = 16 edges. e = edge_attr@Wl + bl ; msg = relu(x[src]+e) ; atomic agg[dst]
__device__ __forceinline__ void stage_tile(int tile, int E,
                                           const int* __restrict__ ei,
                                           const float* __restrict__ ea,
                                           float* sAbuf, int* sSrcBuf, int* sDstBuf,
                                           int tid) {
    const int e0 = tile * 16;
    if (tid < 16)      sSrcBuf[tid]      = ei[e0 + tid];
    else if (tid < 32) sDstBuf[tid - 16] = ei[E + e0 + (tid - 16)];
#pragma unroll
    for (int i = 0; i < 2; ++i) {          // 16 rows x 128 f32 = 512 x b128
        const int f4  = tid + i * 256;
        const int row = f4 >> 5;
        const int c4  = (f4 & 31) * 4;
        async_ld_b128(&sAbuf[row * LDA + c4],
                      ea + (size_t)(e0 + row) * DD + c4);
    }
}

__global__ __launch_bounds__(256)
void k_edge(const float* __restrict__ x, const int* __restrict__ ei,
            const float* __restrict__ ea, const float* __restrict__ Wl,
            const float* __restrict__ bl, float* __restrict__ agg,
            int E, int nTiles)
{
    __shared__ float sA[2][16 * LDA];
    __shared__ int   sSrc[2][16];
    __shared__ int   sDst[2][16];

    const int tid  = threadIdx.x;
    const int lane = tid & 31;
    const int wave = tid >> 5;          // 0..7, each wave owns 16 output cols
    const int half = lane >> 4;         // 0: lanes 0-15, 1: lanes 16-31
    const int l16  = lane & 15;
    const int nCol = wave * 16 + l16;   // 0..127

    // Preload this wave's B slice of Wl: K=128 as 32 steps of K=4.
    // B layout (4x16): VGPR0/1 lanes0-15 = K0/K1, lanes16-31 = K2/K3.
    v2f B[32];
#pragma unroll
    for (int k = 0; k < 32; ++k) {
        const int k0 = 4 * k + 2 * half;
        B[k].x = Wl[(k0    ) * DD + nCol];
        B[k].y = Wl[(k0 + 1) * DD + nCol];
    }
    const float blv = bl[nCol];

    int cur = 0;
    if (blockIdx.x < nTiles)            // block-uniform
        stage_tile(blockIdx.x, E, ei, ea, sA[0], sSrc[0], sDst[0], tid);

    for (int tile = blockIdx.x; tile < nTiles; tile += gridDim.x) {
        wait_async0();                  // our async writes to sA[cur] done
        __syncthreads();                // all waves' writes visible / prev reads done

        // prefetch gather + destination indices for this tile (hides under WMMA)
        int   dsts[8];
        float xg[8];
#pragma unroll
        for (int r = 0; r < 8; ++r) {
            const int m = r + half * 8;
            const int s = sSrc[cur][m];
            dsts[r] = sDst[cur][m];
            xg[r]   = x[(size_t)s * DD + nCol];
        }

        // kick off next tile's DMA into the other buffer
        const int nxt = tile + gridDim.x;
        if (nxt < nTiles)               // block-uniform
            stage_tile(nxt, E, ei, ea, sA[cur ^ 1], sSrc[cur ^ 1], sDst[cur ^ 1], tid);

        // GEMM: e = edge_attr @ Wl
        v8f c = {};
#pragma unroll
        for (int k = 0; k < 32; ++k) {
            const int k0 = 4 * k + 2 * half;
            const float2 av = *(const float2*)&sA[cur][l16 * LDA + k0];
            v2f a; a.x = av.x; a.y = av.y;
            c = __builtin_amdgcn_wmma_f32_16x16x4_f32(
                    false, a, false, B[k], (short)0, c, false, false);
        }

        // epilogue: C layout -> VGPR r holds (M = r + 8*half, N = nCol)
#pragma unroll
        for (int r = 0; r < 8; ++r) {
            float v = c[r] + blv + xg[r];
            v = v > 0.f ? v : 0.f;
            atomicAdd(&agg[(size_t)dsts[r] * DD + nCol], v);
        }
        cur ^= 1;
    }
}

// ---------------------- node MLP + residual + stats ------------------------
__global__ __launch_bounds__(256)
void k_node(const float* __restrict__ x, const float* __restrict__ agg,
            const float* __restrict__ W1, const float* __restrict__ b1,
            const float* __restrict__ W2, const float* __restrict__ b2,
            float* __restrict__ h2, double* __restrict__ sums, int nTiles)
{
    __shared__ float  sH[16 * LDA];
    __shared__ float  sT[16 * LDA];
    __shared__ double wsum[8];
    __shared__ double wsq[8];

    const int tid  = threadIdx.x;
    const int lane = tid & 31;
    const int wave = tid >> 5;
    const int half = lane >> 4;
    const int l16  = lane & 15;
    const int nCol = wave * 16 + l16;

    v2f B1[32], B2[32];
#pragma unroll
    for (int k = 0; k < 32; ++k) {
        const int k0 = 4 * k + 2 * half;
        B1[k].x = W1[(k0    ) * DD + nCol];
        B1[k].y = W1[(k0 + 1) * DD + nCol];
        B2[k].x = W2[(k0    ) * DD + nCol];
        B2[k].y = W2[(k0 + 1) * DD + nCol];
    }
    const float b1v = b1[nCol];
    const float b2v = b2[nCol];

    double lsum = 0.0, lsq = 0.0;

    for (int tile = blockIdx.x; tile < nTiles; tile += gridDim.x) {
        const int n0 = tile * 16;
        // h = x + agg, staged to LDS
#pragma unroll
        for (int i = 0; i < 2; ++i) {
            const int f4  = tid + i * 256;
            const int row = f4 >> 5;
            const int c4  = (f4 & 31) * 4;
            const size_t g = (size_t)(n0 + row) * DD + c4;
            const v4f xv = *(const v4f*)(x   + g);
            const v4f av = *(const v4f*)(agg + g);
            float* d = &sH[row * LDA + c4];
            d[0] = xv[0] + av[0]; d[1] = xv[1] + av[1];
            d[2] = xv[2] + av[2]; d[3] = xv[3] + av[3];
        }
        __syncthreads();

        // GEMM1: t = relu(h @ W1 + b1)
        v8f c = {};
#pragma unroll
        for (int k = 0; k < 32; ++k) {
            const int k0 = 4 * k + 2 * half;
            const float2 av = *(const float2*)&sH[l16 * LDA + k0];
            v2f a; a.x = av.x; a.y = av.y;
            c = __builtin_amdgcn_wmma_f32_16x16x4_f32(
                    false, a, false, B1[k], (short)0, c, false, false);
        }
#pragma unroll
        for (int r = 0; r < 8; ++r) {
            const int m = r + half * 8;
            float v = c[r] + b1v;
            sT[m * LDA + nCol] = v > 0.f ? v : 0.f;
        }
        __syncthreads();

        // GEMM2: h2 = t @ W2 + b2 + x   (residual)
        v8f c2 = {};
#pragma unroll
        for (int k = 0; k < 32; ++k) {
            const int k0 = 4 * k + 2 * half;
            const float2 av = *(const float2*)&sT[l16 * LDA + k0];
            v2f a; a.x = av.x; a.y = av.y;
            c2 = __builtin_amdgcn_wmma_f32_16x16x4_f32(
                    false, a, false, B2[k], (short)0, c2, false, false);
        }
#pragma unroll
        for (int r = 0; r < 8; ++r) {
            const int m = r + half * 8;
            const size_t g = (size_t)(n0 + m) * DD + nCol;
            const float v = c2[r] + b2v + x[g];
            h2[g] = v;
            lsum += (double)v;
            lsq  += (double)v * (double)v;
        }
        __syncthreads();
    }

    // block reduction of LayerNorm statistics
#pragma unroll
    for (int off = 16; off > 0; off >>= 1) {
        lsum += __shfl_down(lsum, off, 32);
        lsq  += __shfl_down(lsq,  off, 32);
    }
    if (lane == 0) { wsum[wave] = lsum; wsq[wave] = lsq; }
    __syncthreads();
    if (tid == 0) {
        double a = 0.0, b = 0.0;
#pragma unroll
        for (int w = 0; w < 8; ++w) { a += wsum[w]; b += wsq[w]; }
        atomicAdd(&sums[0], a);
        atomicAdd(&sums[1], b);
    }
}

// -------------------- normalize + affine + SiLU + nan_to_num ---------------
__global__ void k_norm(const float* __restrict__ h2, const double* __restrict__ sums,
                       const float* __restrict__ lnw, const float* __restrict__ lnb,
                       float* __restrict__ out, int total4)
{
    const double cnt = (double)total4 * 4.0;
    const double mu_d = sums[0] / cnt;
    const double var_d = sums[1] / cnt - mu_d * mu_d;
    const float mu    = (float)mu_d;
    const float stdv  = sqrtf(fmaxf((float)var_d, 0.f));
    const float scale = 1.f / (stdv + 1e-5f);

    for (int i = blockIdx.x * blockDim.x + threadIdx.x; i < total4;
         i += gridDim.x * blockDim.x) {
        v4f v = ((const v4f*)h2)[i];
        const int c4 = (i & 31) * 4;          // 32 float4 per 128-wide row
        v4f o;
#pragma unroll
        for (int j = 0; j < 4; ++j) {
            float t = (v[j] - mu) * scale * lnw[c4 + j] + lnb[c4 + j];
            t = t / (1.f + __expf(-t));        // SiLU
            if (isnan(t)) t = 0.f;             // nan_to_num
            else if (isinf(t)) t = t > 0.f ? FLT_MAX : -FLT_MAX;
            o[j] = t;
        }
        ((v4f*)out)[i] = o;
    }
}

// ---------------------------------------------------------------------------
extern "C" void kernel_launch(void* const* d_in, const int* in_sizes, int n_in,
                              void* d_out, int out_size, void* d_ws, size_t ws_size,
                              hipStream_t stream) {
    const float* x   = (const float*)d_in[0];
    const int*   ei  = (const int*)  d_in[1];   // [2,E]: src then dst
    const float* ea  = (const float*)d_in[2];
    const float* bl  = (const float*)d_in[4];
    const float* Wl  = (const float*)d_in[3];
    const float* W1  = (const float*)d_in[5];
    const float* b1  = (const float*)d_in[6];
    const float* W2  = (const float*)d_in[7];
    const float* b2  = (const float*)d_in[8];
    const float* lnw = (const float*)d_in[9];
    const float* lnb = (const float*)d_in[10];

    const int N = in_sizes[0] / DD;
    const int E = in_sizes[2] / DD;
    const int total4 = (N * DD) / 4;

    float*  agg  = (float*)d_out;                       // reused as agg buffer
    float*  h2   = (float*)d_ws;                        // N*D floats
    double* sums = (double*)((char*)d_ws + (size_t)N * DD * sizeof(float));

    k_zero<<<1024, 256, 0, stream>>>(agg, sums, total4);
    k_edge<<<2048, 256, 0, stream>>>(x, ei, ea, Wl, bl, agg, E, E / 16);
    k_node<<<1024, 256, 0, stream>>>(x, agg, W1, b1, W2, b2, h2, sums, N / 16);
    k_norm<<<2048, 256, 0, stream>>>(h2, sums, lnw, lnb, (float*)d_out, total4);
}